// CGClass_90271622628138
// MI455X (gfx1250) — compile-verified
//
#include <hip/hip_runtime.h>
#include <math.h>

#define HN 320
#define HW (HN*HN)
#define NB 8
#define NC 12
#define NIMG (NB*NC)
#define CGTOL 1e-10f
#define MAX_ITER 10

#define LDSPITCH 72            // 64 cols + 8 pad -> half-waves hit disjoint banks
#define IMOFF (32*LDSPITCH)    // im-plane offset inside LDS slab

typedef __attribute__((ext_vector_type(2))) float v2f;
typedef __attribute__((ext_vector_type(8))) float v8f;

// D = A(16x4 f32) * B(4x16 f32) + C(16x16 f32), wave32 WMMA
__device__ __forceinline__ v8f wmma4(v2f a, v2f b, v8f c) {
  return __builtin_amdgcn_wmma_f32_16x16x4_f32(
      /*neg_a=*/false, a, /*neg_b=*/false, b,
      /*c_mod=*/(short)0, c, /*reuse_a=*/false, /*reuse_b=*/false);
}

// ---------------------------------------------------------------------------
// DFT matrix generation: F = exp(-i*2pi*jk/N)/sqrt(N) (fwd), G = conj(F) (inv)
// ---------------------------------------------------------------------------
__global__ void k_gen_dft(float* __restrict__ Fre, float* __restrict__ Fim,
                          float* __restrict__ Gre, float* __restrict__ Gim) {
  int i = blockIdx.x * blockDim.x + threadIdx.x;
  if (i >= HW) return;
  int j = i / HN, k = i - j * HN;
  int t = (j * k) % HN;                       // exact phase index
  float ang = (6.2831853071795864769f / HN) * (float)t;
  float c = cosf(ang), s = sinf(ang);
  const float r = 0.05590169943749474f;       // 1/sqrt(320)
  Fre[i] = c * r;  Fim[i] = -s * r;
  Gre[i] = c * r;  Gim[i] =  s * r;
}

// ---------------------------------------------------------------------------
// Batched complex GEMM: C_img = A_img(320x320) * B_img(320x320)
// 128-thread blocks (4 waves). Block -> 64x64 output region; wave -> 16x64
// strip (4 N-tiles). B slab (32 K-rows x 64 cols, re+im) staged in LDS with
// padded pitch; A fragment reused across the 4 tiles from registers.
// ---------------------------------------------------------------------------
__global__ void __launch_bounds__(128) k_cgemm(
    const float* __restrict__ Are, const float* __restrict__ Aim, long aStride,
    const float* __restrict__ Bre, const float* __restrict__ Bim, long bStride,
    float* __restrict__ Cre, float* __restrict__ Cim)
{
  __shared__ float ldsB[2 * 32 * LDSPITCH];   // 18 KB

  const int img  = blockIdx.y;
  const int tid  = threadIdx.x;
  const int wave = tid >> 5;
  const int lane = tid & 31;
  const int bm = blockIdx.x / 5;              // 0..4 -> 64 M-rows each
  const int bn = blockIdx.x - bm * 5;         // 0..4 -> 64 N-cols each
  const int h  = lane >> 4;                   // half: K pair select
  const int l  = lane & 15;
  const int m0 = bm * 64 + wave * 16;         // wave's M strip
  const int n0 = bn * 64;                     // block's N base
  const int m  = m0 + l;

  const float* are = Are + (long)img * aStride;
  const float* aim = Aim + (long)img * aStride;
  const float* bre = Bre + (long)img * bStride;
  const float* bim = Bim + (long)img * bStride;

  // staging coords: 8 row-groups x 16 col-groups of float4
  const int rg = tid >> 4;                    // 0..7
  const int c4 = (tid & 15) * 4;              // 0..60

  v8f rr[4], ii[4], ri[4], ir[4];
#pragma unroll
  for (int t = 0; t < 4; ++t) { rr[t] = (v8f){}; ii[t] = (v8f){}; ri[t] = (v8f){}; ir[t] = (v8f){}; }

  for (int kc = 0; kc < HN; kc += 32) {
    __syncthreads();
    // ---- stage B slab [kc..kc+32) x [n0..n0+64), both planes, into LDS ----
#pragma unroll
    for (int r0 = 0; r0 < 32; r0 += 8) {
      int r = r0 + rg;
      float4 vR = *(const float4*)(bre + (long)(kc + r) * HN + n0 + c4);
      float4 vI = *(const float4*)(bim + (long)(kc + r) * HN + n0 + c4);
      *(float4*)(ldsB + r * LDSPITCH + c4)         = vR;
      *(float4*)(ldsB + IMOFF + r * LDSPITCH + c4) = vI;
    }
    if (kc + 32 < HN) {                       // uniform guard: prefetch next slab
      __builtin_prefetch(bre + (long)(kc + 32 + rg) * HN + n0 + c4, 0, 3);
      __builtin_prefetch(bim + (long)(kc + 32 + rg) * HN + n0 + c4, 0, 3);
    }
    __syncthreads();

    // ---- compute: 8 K-steps of 4, 16 WMMAs per step ----
#pragma unroll
    for (int k0 = 0; k0 < 32; k0 += 4) {
      const int ka = k0 + 2 * h;              // local K pair for this half
      v2f aR = *(const v2f*)(are + (long)m * HN + kc + ka);
      v2f aI = *(const v2f*)(aim + (long)m * HN + kc + ka);
#pragma unroll
      for (int t = 0; t < 4; ++t) {
        const int col = t * 16 + l;
        v2f bR; bR.x = ldsB[ka * LDSPITCH + col];
                bR.y = ldsB[(ka + 1) * LDSPITCH + col];
        v2f bI; bI.x = ldsB[IMOFF + ka * LDSPITCH + col];
                bI.y = ldsB[IMOFF + (ka + 1) * LDSPITCH + col];
        rr[t] = wmma4(aR, bR, rr[t]);         // Re*Re
        ii[t] = wmma4(aI, bI, ii[t]);         // Im*Im
        ri[t] = wmma4(aR, bI, ri[t]);         // Re*Im
        ir[t] = wmma4(aI, bR, ir[t]);         // Im*Re
      }
    }
  }

  float* cre = Cre + (long)img * HW;
  float* cim = Cim + (long)img * HW;
#pragma unroll
  for (int t = 0; t < 4; ++t) {
    const int col = n0 + t * 16 + l;
#pragma unroll
    for (int i = 0; i < 8; ++i) {             // D layout: VGPR i -> rows i, i+8
      int row = m0 + h * 8 + i;
      cre[row * HN + col] = rr[t][i] - ii[t][i];
      cim[row * HN + col] = ri[t][i] + ir[t][i];
    }
  }
}

// ---------------------------------------------------------------------------
// Elementwise kernels
// ---------------------------------------------------------------------------
__global__ void k_mulS(const float* __restrict__ sre, const float* __restrict__ sim,
                       const float* __restrict__ pre, const float* __restrict__ pim,
                       float* __restrict__ ore, float* __restrict__ oim) {
  int i = blockIdx.x * blockDim.x + threadIdx.x;
  if (i >= NIMG * HW) return;
  int img = i / HW, pix = i - img * HW;
  int b = img / NC;
  float sr = sre[i], si = sim[i];
  float pr = pre[b * HW + pix], pi = pim[b * HW + pix];
  ore[i] = sr * pr - si * pi;
  oim[i] = sr * pi + si * pr;
}

__global__ void k_masky(const float* __restrict__ mask,
                        const float* __restrict__ yre, const float* __restrict__ yim,
                        float* __restrict__ ore, float* __restrict__ oim) {
  int i = blockIdx.x * blockDim.x + threadIdx.x;
  if (i >= NIMG * HW) return;
  int img = i / HW, pix = i - img * HW;
  int b = img / NC;
  float mv = mask[b * HW + pix];
  ore[i] = mv * yre[i];
  oim[i] = mv * yim[i];
}

__global__ void k_mask2(const float* __restrict__ mask,
                        float* __restrict__ re, float* __restrict__ im) {
  int i = blockIdx.x * blockDim.x + threadIdx.x;
  if (i >= NIMG * HW) return;
  int img = i / HW, pix = i - img * HW;
  int b = img / NC;
  float mv = mask[b * HW + pix];
  mv *= mv;
  re[i] *= mv;
  im[i] *= mv;
}

// out_b = sum_c conj(S_c) * Z_c + lam * add_b
__global__ void k_reduceS(const float* __restrict__ sre, const float* __restrict__ sim,
                          const float* __restrict__ zre, const float* __restrict__ zim,
                          const float* __restrict__ addRe, const float* __restrict__ addIm,
                          const float* __restrict__ lam,
                          float* __restrict__ ore, float* __restrict__ oim) {
  int i = blockIdx.x * blockDim.x + threadIdx.x;
  if (i >= NB * HW) return;
  int b = i / HW, pix = i - b * HW;
  float ar = 0.f, ai = 0.f;
  for (int c = 0; c < NC; ++c) {
    int idx = (b * NC + c) * HW + pix;
    float sr = sre[idx], si = sim[idx];
    float zr = zre[idx], zi = zim[idx];
    ar += sr * zr + si * zi;                  // Re(conj(s)*z)
    ai += sr * zi - si * zr;                  // Im(conj(s)*z)
  }
  float L = lam[0];
  ore[i] = ar + L * addRe[i];
  oim[i] = ai + L * addIm[i];
}

__global__ void k_init(const float* __restrict__ rhsRe, const float* __restrict__ rhsIm,
                       float* __restrict__ xre, float* __restrict__ xim,
                       float* __restrict__ rre, float* __restrict__ rim,
                       float* __restrict__ pre, float* __restrict__ pim) {
  int i = blockIdx.x * blockDim.x + threadIdx.x;
  if (i >= NB * HW) return;
  float vr = rhsRe[i], vi = rhsIm[i];
  xre[i] = 0.f; xim[i] = 0.f;
  rre[i] = vr;  rim[i] = vi;
  pre[i] = vr;  pim[i] = vi;
}

// ---------------------------------------------------------------------------
// Deterministic per-batch reductions: Re(<a,b>) = sum aRe*bRe + aIm*bIm
// ---------------------------------------------------------------------------
__global__ void __launch_bounds__(256) k_dot1(
    const float* __restrict__ aRe, const float* __restrict__ aIm,
    const float* __restrict__ bRe, const float* __restrict__ bIm,
    float* __restrict__ partial) {
  __shared__ float sm[256];
  int b = blockIdx.y;
  int base = b * HW;
  int pix = blockIdx.x * 512 + threadIdx.x;
  float s = aRe[base + pix] * bRe[base + pix] + aIm[base + pix] * bIm[base + pix];
  int pix2 = pix + 256;
  s += aRe[base + pix2] * bRe[base + pix2] + aIm[base + pix2] * bIm[base + pix2];
  sm[threadIdx.x] = s;
  __syncthreads();
  for (int off = 128; off > 0; off >>= 1) {
    if (threadIdx.x < off) sm[threadIdx.x] += sm[threadIdx.x + off];
    __syncthreads();
  }
  if (threadIdx.x == 0) partial[b * 200 + blockIdx.x] = sm[0];
}

__global__ void k_dot2(const float* __restrict__ partial, float* __restrict__ out) {
  int b = threadIdx.x;
  if (b >= NB) return;
  float s = 0.f;
  for (int j = 0; j < 200; ++j) s += partial[b * 200 + j];
  out[b] = s;
}

// ---------------------------------------------------------------------------
// CG updates with per-batch active gating (rTr > tol)
// ---------------------------------------------------------------------------
__global__ void k_update_xr(const float* __restrict__ rTr, const float* __restrict__ pAp,
                            const float* __restrict__ pre, const float* __restrict__ pim,
                            const float* __restrict__ mre, const float* __restrict__ mim,
                            float* __restrict__ xre, float* __restrict__ xim,
                            float* __restrict__ rre, float* __restrict__ rim) {
  int i = blockIdx.x * blockDim.x + threadIdx.x;
  if (i >= NB * HW) return;
  int b = i / HW;
  float rtr = rTr[b];
  if (!(rtr > CGTOL)) return;
  float alpha = rtr / pAp[b];
  xre[i] += alpha * pre[i];
  xim[i] += alpha * pim[i];
  rre[i] -= alpha * mre[i];
  rim[i] -= alpha * mim[i];
}

__global__ void k_update_p(const float* __restrict__ rTr, const float* __restrict__ rTrN,
                           const float* __restrict__ rre, const float* __restrict__ rim,
                           float* __restrict__ pre, float* __restrict__ pim) {
  int i = blockIdx.x * blockDim.x + threadIdx.x;
  if (i >= NB * HW) return;
  int b = i / HW;
  float rtr = rTr[b];
  if (!(rtr > CGTOL)) return;
  float beta = rTrN[b] / rtr;
  pre[i] = rre[i] + beta * pre[i];
  pim[i] = rim[i] + beta * pim[i];
}

__global__ void k_update_rtr(float* __restrict__ rTr, const float* __restrict__ rTrN) {
  int b = threadIdx.x;
  if (b >= NB) return;
  float v = rTr[b];
  rTr[b] = (v > CGTOL) ? rTrN[b] : v;
}

__global__ void k_writeout(const float* __restrict__ xre, const float* __restrict__ xim,
                           float* __restrict__ out) {
  int i = blockIdx.x * blockDim.x + threadIdx.x;
  if (i >= NB * HW) return;
  out[2 * i]     = xre[i];
  out[2 * i + 1] = xim[i];
}

// ---------------------------------------------------------------------------
extern "C" void kernel_launch(void* const* d_in, const int* in_sizes, int n_in,
                              void* d_out, int out_size, void* d_ws, size_t ws_size,
                              hipStream_t stream) {
  const float* lam  = (const float*)d_in[0];
  const float* x_re = (const float*)d_in[1];
  const float* x_im = (const float*)d_in[2];
  const float* y_re = (const float*)d_in[3];
  const float* y_im = (const float*)d_in[4];
  const float* s_re = (const float*)d_in[5];
  const float* s_im = (const float*)d_in[6];
  const float* mask = (const float*)d_in[7];

  // Workspace carving (~192 MB of f32)
  float* w = (float*)d_ws;
  float* Fre = w; w += HW;                    // forward DFT matrix
  float* Fim = w; w += HW;
  float* Gre = w; w += HW;                    // inverse DFT matrix (conj)
  float* Gim = w; w += HW;
  float* b0re = w; w += (long)NIMG * HW;      // ping-pong image buffers
  float* b0im = w; w += (long)NIMG * HW;
  float* b1re = w; w += (long)NIMG * HW;
  float* b1im = w; w += (long)NIMG * HW;
  float* rhsRe = w; w += NB * HW;
  float* rhsIm = w; w += NB * HW;
  float* xre = w; w += NB * HW;
  float* xim = w; w += NB * HW;
  float* rre = w; w += NB * HW;
  float* rim = w; w += NB * HW;
  float* pre = w; w += NB * HW;
  float* pim = w; w += NB * HW;
  float* mpre = w; w += NB * HW;
  float* mpim = w; w += NB * HW;
  float* rTr  = w; w += NB;
  float* pAp  = w; w += NB;
  float* rTrN = w; w += NB;
  float* part = w; w += NB * 200;

  const dim3 gGemm(25, NIMG);                 // 5x5 blocks of 64x64 per image
  const int  bGemm = 128;                     // 4 waves/block
  const int gImg = (NIMG * HW + 255) / 256;   // 38400
  const int gB   = (NB * HW + 255) / 256;     // 3200
  const dim3 gDot(200, NB);

  // ---- setup: rhs = AH(y) + lam*x; x=0, r=p=rhs; rTr = |r|^2 ----
  k_gen_dft<<<(HW + 255) / 256, 256, 0, stream>>>(Fre, Fim, Gre, Gim);
  k_masky<<<gImg, 256, 0, stream>>>(mask, y_re, y_im, b0re, b0im);
  k_cgemm<<<gGemm, bGemm, 0, stream>>>(Gre, Gim, 0, b0re, b0im, (long)HW, b1re, b1im);
  k_cgemm<<<gGemm, bGemm, 0, stream>>>(b1re, b1im, (long)HW, Gre, Gim, 0, b0re, b0im);
  k_reduceS<<<gB, 256, 0, stream>>>(s_re, s_im, b0re, b0im, x_re, x_im, lam, rhsRe, rhsIm);
  k_init<<<gB, 256, 0, stream>>>(rhsRe, rhsIm, xre, xim, rre, rim, pre, pim);
  k_dot1<<<gDot, 256, 0, stream>>>(rre, rim, rre, rim, part);
  k_dot2<<<1, 32, 0, stream>>>(part, rTr);

  // ---- CG iterations ----
  for (int it = 0; it < MAX_ITER; ++it) {
    // Ap = AH(A(p)) + lam*p
    k_mulS<<<gImg, 256, 0, stream>>>(s_re, s_im, pre, pim, b0re, b0im);
    k_cgemm<<<gGemm, bGemm, 0, stream>>>(Fre, Fim, 0, b0re, b0im, (long)HW, b1re, b1im);
    k_cgemm<<<gGemm, bGemm, 0, stream>>>(b1re, b1im, (long)HW, Fre, Fim, 0, b0re, b0im);
    k_mask2<<<gImg, 256, 0, stream>>>(mask, b0re, b0im);
    k_cgemm<<<gGemm, bGemm, 0, stream>>>(Gre, Gim, 0, b0re, b0im, (long)HW, b1re, b1im);
    k_cgemm<<<gGemm, bGemm, 0, stream>>>(b1re, b1im, (long)HW, Gre, Gim, 0, b0re, b0im);
    k_reduceS<<<gB, 256, 0, stream>>>(s_re, s_im, b0re, b0im, pre, pim, lam, mpre, mpim);
    // pAp = Re(<p, Ap>)
    k_dot1<<<gDot, 256, 0, stream>>>(pre, pim, mpre, mpim, part);
    k_dot2<<<1, 32, 0, stream>>>(part, pAp);
    // x += alpha p; r -= alpha Ap   (gated on rTr > tol)
    k_update_xr<<<gB, 256, 0, stream>>>(rTr, pAp, pre, pim, mpre, mpim, xre, xim, rre, rim);
    // rTrNew = |r|^2
    k_dot1<<<gDot, 256, 0, stream>>>(rre, rim, rre, rim, part);
    k_dot2<<<1, 32, 0, stream>>>(part, rTrN);
    // p = r + beta p; rTr <- rTrNew (gated)
    k_update_p<<<gB, 256, 0, stream>>>(rTr, rTrN, rre, rim, pre, pim);
    k_update_rtr<<<1, 32, 0, stream>>>(rTr, rTrN);
  }

  k_writeout<<<gB, 256, 0, stream>>>(xre, xim, (float*)d_out);
}